// STAdaptor_36404142801577
// MI455X (gfx1250) — compile-verified
//
#include <hip/hip_runtime.h>
#include <math.h>

typedef __attribute__((ext_vector_type(2))) float v2f;
typedef __attribute__((ext_vector_type(8))) float v8f;

#define EPSN 1e-12f
#define E_ 2
#define N_ 512
#define B_ 64
#define T_ 12
#define K_ 64
#define TPAD 16

__device__ __forceinline__ float softplusf(float x) {
    return (x > 30.f) ? x : log1pf(expf(x));
}

// ---------------------------------------------------------------------------
// Kernel 1: sup[e,n,m] = supports[e,n,m] + sum_k u[e,n,k]*ss[e,k]*v[e,m,k]
// One wave (32 threads) per 16x16 output tile; 16 k-steps of f32 WMMA (K=64).
// ---------------------------------------------------------------------------
__global__ __launch_bounds__(32)
void sup_wmma_kernel(const float* __restrict__ supports,
                     const float* __restrict__ u,
                     const float* __restrict__ s,
                     const float* __restrict__ v,
                     const float* __restrict__ bias,
                     float* __restrict__ sup) {
    const int e    = blockIdx.z;
    const int n0   = blockIdx.y * 16;
    const int m0   = blockIdx.x * 16;
    const int lane = threadIdx.x & 31;
    const int half = lane >> 4;     // 0: K pairs {k0,k0+1}, 1: {k0+2,k0+3}
    const int l16  = lane & 15;

    __shared__ float ss[K_];
    ss[lane]      = s[e * K_ + lane]      * softplusf(bias[e * K_ + lane]);
    ss[lane + 32] = s[e * K_ + lane + 32] * softplusf(bias[e * K_ + lane + 32]);
    __syncthreads();

    const float* ue = u + (size_t)e * N_ * K_;
    const float* ve = v + (size_t)e * N_ * K_;

    v8f acc = {};
#pragma unroll
    for (int kk = 0; kk < 16; ++kk) {
        const int k = kk * 4 + half * 2;
        // A-frag: rows = n (M), scaled by singular values
        const float* pa = ue + (size_t)(n0 + l16) * K_ + k;
        v2f a;
        a.x = pa[0] * ss[k];
        a.y = pa[1] * ss[k + 1];
        // B-frag: B[k][m] = v[m][k]  (columns = m)
        const float* pb = ve + (size_t)(m0 + l16) * K_ + k;
        v2f b;
        b.x = pb[0];
        b.y = pb[1];
        acc = __builtin_amdgcn_wmma_f32_16x16x4_f32(false, a, false, b,
                                                    (short)0, acc, false, false);
    }

    const float* se = supports + (size_t)e * N_ * N_;
    float*       oe = sup      + (size_t)e * N_ * N_;
#pragma unroll
    for (int r = 0; r < 8; ++r) {
        const int row = n0 + r + 8 * half;   // C/D layout: VGPR r -> M = r + 8*half
        const int col = m0 + l16;
        oe[(size_t)row * N_ + col] = acc[r] + se[(size_t)row * N_ + col];
    }
}

// ---------------------------------------------------------------------------
// Kernel 2: nodes[b,n,0:12] = normalize(inputs[b,:,n,0]); pad k=12..15 with 0.
// Also store sq[b,n] = sum(nodes^2) (matches reference's sq on normalized x).
// ---------------------------------------------------------------------------
__global__ __launch_bounds__(256)
void normalize_kernel(const float* __restrict__ inputs,
                      float* __restrict__ nodes,
                      float* __restrict__ sq) {
    const int idx = blockIdx.x * blockDim.x + threadIdx.x;  // b*N + n
    if (idx >= B_ * N_) return;
    const int b = idx / N_;
    const int n = idx % N_;

    float x[T_];
    float s2 = 0.f;
#pragma unroll
    for (int t = 0; t < T_; ++t) {
        x[t] = inputs[((size_t)(b * T_ + t) * N_ + n) * 2];  // Fdim=2, f=0
        s2 += x[t] * x[t];
    }
    const float inv = 1.f / fmaxf(sqrtf(s2), EPSN);

    float q = 0.f;
    float* np = nodes + (size_t)idx * TPAD;
#pragma unroll
    for (int t = 0; t < T_; ++t) {
        const float y = x[t] * inv;
        np[t] = y;
        q += y * y;
    }
#pragma unroll
    for (int t = T_; t < TPAD; ++t) np[t] = 0.f;
    sq[idx] = q;
}

// ---------------------------------------------------------------------------
// Kernel 3 (fused, HBM-optimal): per (b, 16-row strip):
//   phase 1: WMMA pairwise inner products -> (exp(-dist)+1) strip in LDS
//   phase 2: for e=0,1: relu(sup*(dyn+1)), wave32 row-sum, L1 normalize, store
// ---------------------------------------------------------------------------
__global__ __launch_bounds__(256)
void fused_out_kernel(const float* __restrict__ sup,
                      const float* __restrict__ nodes,
                      const float* __restrict__ sqv,
                      float* __restrict__ out) {
    const int b    = blockIdx.y;
    const int n0   = blockIdx.x * 16;
    const int tid  = threadIdx.x;
    const int wave = tid >> 5;
    const int lane = tid & 31;
    const int half = lane >> 4;
    const int l16  = lane & 15;

    __shared__ float dyn[16 * N_];   // 32 KB: (exp(-dist)+1) for 16 rows x 512 cols

    const float* nb  = nodes + (size_t)b * N_ * TPAD;
    const float* sqb = sqv   + (size_t)b * N_;

    // -------- phase 1: 32 m-tiles across 8 waves (4 each) --------
    for (int j = wave; j < N_ / 16; j += 8) {
        const int m0 = j * 16;
        v8f acc = {};
#pragma unroll
        for (int kk = 0; kk < 4; ++kk) {   // K = 16 (12 real + 4 zero pad)
            const int k = kk * 4 + half * 2;
            const float* pa = nb + (size_t)(n0 + l16) * TPAD + k;
            v2f a;  a.x = pa[0]; a.y = pa[1];
            const float* pb = nb + (size_t)(m0 + l16) * TPAD + k;
            v2f bb; bb.x = pb[0]; bb.y = pb[1];
            acc = __builtin_amdgcn_wmma_f32_16x16x4_f32(false, a, false, bb,
                                                        (short)0, acc, false, false);
        }
        const float sqm = sqb[m0 + l16];
#pragma unroll
        for (int r = 0; r < 8; ++r) {
            const int row = r + 8 * half;
            float d2 = fmaxf(sqb[n0 + row] + sqm - 2.f * acc[r], 0.f);
            const float dist = (d2 > 0.f) ? sqrtf(d2) : 0.f;
            dyn[row * N_ + m0 + l16] = expf(-dist) + 1.0f;
        }
    }
    __syncthreads();

    // -------- phase 2: each wave normalizes 2 rows, both e --------
    for (int rr = 0; rr < 2; ++rr) {
        const int row = wave * 2 + rr;
        const int n   = n0 + row;
        for (int e = 0; e < E_; ++e) {
            const float* sp = sup + ((size_t)e * N_ + n) * N_;
            float4 vals[4];
            float sum = 0.f;
#pragma unroll
            for (int i = 0; i < 4; ++i) {
                const int m = lane * 4 + i * 128;
                const float4 dv = *(const float4*)&dyn[row * N_ + m];
                const float4 sv = *(const float4*)&sp[m];
                float4 r4;
                r4.x = fmaxf(sv.x * dv.x, 0.f);
                r4.y = fmaxf(sv.y * dv.y, 0.f);
                r4.z = fmaxf(sv.z * dv.z, 0.f);
                r4.w = fmaxf(sv.w * dv.w, 0.f);
                vals[i] = r4;
                sum += r4.x + r4.y + r4.z + r4.w;
            }
#pragma unroll
            for (int off = 16; off >= 1; off >>= 1)
                sum += __shfl_xor(sum, off, 32);
            const float inv = 1.f / fmaxf(sum, EPSN);

            float* op = out + (((size_t)e * B_ + b) * N_ + n) * N_;
#pragma unroll
            for (int i = 0; i < 4; ++i) {
                const int m = lane * 4 + i * 128;
                float4 r4 = vals[i];
                r4.x *= inv; r4.y *= inv; r4.z *= inv; r4.w *= inv;
                *(float4*)&op[m] = r4;
            }
        }
    }
}

// ---------------------------------------------------------------------------
extern "C" void kernel_launch(void* const* d_in, const int* in_sizes, int n_in,
                              void* d_out, int out_size, void* d_ws, size_t ws_size,
                              hipStream_t stream) {
    (void)in_sizes; (void)n_in; (void)out_size; (void)ws_size;
    const float* supports = (const float*)d_in[0];
    const float* u        = (const float*)d_in[1];
    const float* s        = (const float*)d_in[2];
    const float* v        = (const float*)d_in[3];
    const float* bias     = (const float*)d_in[4];
    const float* inputs   = (const float*)d_in[5];
    float* out = (float*)d_out;

    float* ws    = (float*)d_ws;
    float* sup   = ws;                                  // E*N*N       = 524288 f
    float* nodes = sup + (size_t)E_ * N_ * N_;          // B*N*TPAD    = 524288 f
    float* sq    = nodes + (size_t)B_ * N_ * TPAD;      // B*N         = 32768 f

    sup_wmma_kernel<<<dim3(N_ / 16, N_ / 16, E_), 32, 0, stream>>>(
        supports, u, s, v, bias, sup);
    normalize_kernel<<<(B_ * N_ + 255) / 256, 256, 0, stream>>>(inputs, nodes, sq);
    fused_out_kernel<<<dim3(N_ / 16, B_), 256, 0, stream>>>(sup, nodes, sq, out);
}